// SampledSoftmaxLayer_59485297050156
// MI455X (gfx1250) — compile-verified
//
#include <hip/hip_runtime.h>

#define VOCAB   1000000
#define NSAMP   8192
#define BATCH   4096
#define EMB     64
#define WAVES   8          // 256 threads / 32
#define ROWS_PER_WG 16

typedef float v2f __attribute__((ext_vector_type(2)));
typedef float v8f __attribute__((ext_vector_type(8)));

// log(NSAMP * (log(c+2)-log(c+1)) / log(VOCAB+1))
__device__ __forceinline__ float log_q(int id) {
    const float INV_LOG_V = 0.07238241985f;   // 1 / ln(1000001)
    float c = (float)id;
    float p = (__logf(c + 2.0f) - __logf(c + 1.0f)) * INV_LOG_V;
    return __logf((float)NSAMP * p);
}

__device__ __forceinline__ v8f wmma_f32(v2f a, v2f b, v8f c) {
    return __builtin_amdgcn_wmma_f32_16x16x4_f32(
        /*neg_a=*/false, a, /*neg_b=*/false, b,
        /*c_mod=*/(short)0, c, /*reuse_a=*/false, /*reuse_b=*/false);
}

__global__ __launch_bounds__(256) void sampled_softmax_wmma(
    const float* __restrict__ item_emb,   // [VOCAB, 64]
    const float* __restrict__ user_emb,   // [BATCH, 64]
    const float* __restrict__ zero_bias,  // [VOCAB]
    const int*   __restrict__ label_index,// [BATCH, 1]
    const int*   __restrict__ sampled_ids,// [NSAMP]
    float*       __restrict__ out)        // [BATCH, 1]
{
    __shared__ int   s_labels[ROWS_PER_WG];
    __shared__ float s_true[ROWS_PER_WG];
    __shared__ float s_redM[WAVES][ROWS_PER_WG];
    __shared__ float s_redS[WAVES][ROWS_PER_WG];

    const int m0   = blockIdx.x * ROWS_PER_WG;
    const int tid  = threadIdx.x;
    const int wave = tid >> 5;
    const int lane = tid & 31;
    const int half = lane >> 4;   // 0: lanes 0-15, 1: lanes 16-31
    const int l16  = lane & 15;

    // ---- true logits (16 rows, trivial dot products) ----
    if (tid < ROWS_PER_WG) {
        int lab = label_index[m0 + tid];
        const float* ur = user_emb + (size_t)(m0 + tid) * EMB;
        const float* iw = item_emb + (size_t)lab * EMB;
        float acc = 0.0f;
        #pragma unroll 8
        for (int k = 0; k < EMB; ++k) acc = fmaf(ur[k], iw[k], acc);
        s_true[tid]   = acc + zero_bias[lab] - log_q(lab);
        s_labels[tid] = lab;
    }
    __syncthreads();

    // ---- A fragments (16x4 f32): lane holds M=l16, K = 2*half + v + 4*kc ----
    v2f a_frag[16];
    {
        const float* ar = user_emb + (size_t)(m0 + l16) * EMB + half * 2;
        #pragma unroll
        for (int kc = 0; kc < 16; ++kc)
            a_frag[kc] = *(const v2f*)(ar + kc * 4);
    }

    int lab8[8];
    #pragma unroll
    for (int v = 0; v < 8; ++v) lab8[v] = s_labels[v + 8 * half];

    float accM[8], accS[8];
    #pragma unroll
    for (int v = 0; v < 8; ++v) { accM[v] = -INFINITY; accS[v] = 0.0f; }

    const v8f zero8 = {0.f,0.f,0.f,0.f,0.f,0.f,0.f,0.f};

    // ---- column loop: each wave takes two 16-wide tiles per iteration ----
    for (int n0 = wave * 32; n0 < NSAMP; n0 += WAVES * 32) {
        const int col0 = n0 + l16;
        const int col1 = col0 + 16;
        const int sid0 = sampled_ids[col0];
        const int sid1 = sampled_ids[col1];
        const float* br0 = item_emb + (size_t)sid0 * EMB + half * 2;
        const float* br1 = item_emb + (size_t)sid1 * EMB + half * 2;

        // 4 independent WMMA chains: 2 tiles x 2 kc-parity partials
        v8f c0a = zero8, c0b = zero8, c1a = zero8, c1b = zero8;
        #pragma unroll
        for (int kc = 0; kc < 16; kc += 2) {
            v2f b00 = *(const v2f*)(br0 + kc * 4);
            v2f b01 = *(const v2f*)(br0 + kc * 4 + 4);
            v2f b10 = *(const v2f*)(br1 + kc * 4);
            v2f b11 = *(const v2f*)(br1 + kc * 4 + 4);
            c0a = wmma_f32(a_frag[kc],     b00, c0a);
            c1a = wmma_f32(a_frag[kc],     b10, c1a);
            c0b = wmma_f32(a_frag[kc + 1], b01, c0b);
            c1b = wmma_f32(a_frag[kc + 1], b11, c1b);
        }
        v8f c0 = c0a + c0b;
        v8f c1 = c1a + c1b;

        // branchless prefetch of next iteration's gathered rows
        {
            int nc0 = col0 + WAVES * 32; nc0 = (nc0 < NSAMP) ? nc0 : col0;
            int nc1 = col1 + WAVES * 32; nc1 = (nc1 < NSAMP) ? nc1 : col1;
            int ns0 = sampled_ids[nc0];
            int ns1 = sampled_ids[nc1];
            __builtin_prefetch(item_emb + (size_t)ns0 * EMB + half * 32, 0, 1);
            __builtin_prefetch(item_emb + (size_t)ns1 * EMB + half * 32, 0, 1);
        }

        const float addt0 = zero_bias[sid0] - log_q(sid0);
        const float addt1 = zero_bias[sid1] - log_q(sid1);
        #pragma unroll
        for (int v = 0; v < 8; ++v) {
            float val = c0[v] + addt0;              // row (v+8*half), col0
            if (lab8[v] == sid0) val = -1.0e9f;     // hit mask
            float d = val - accM[v];                // online logsumexp
            if (d > 0.0f) { accS[v] = accS[v] * __expf(-d) + 1.0f; accM[v] = val; }
            else          { accS[v] += __expf(d); }

            val = c1[v] + addt1;                    // row (v+8*half), col1
            if (lab8[v] == sid1) val = -1.0e9f;
            d = val - accM[v];
            if (d > 0.0f) { accS[v] = accS[v] * __expf(-d) + 1.0f; accM[v] = val; }
            else          { accS[v] += __expf(d); }
        }
    }

    // ---- reduce across the 16 lanes of each half (cols live on l16) ----
    #pragma unroll
    for (int v = 0; v < 8; ++v) {
        #pragma unroll
        for (int off = 1; off < 16; off <<= 1) {
            float oM = __shfl_xor(accM[v], off, 32);
            float oS = __shfl_xor(accS[v], off, 32);
            float nM = fmaxf(accM[v], oM);
            accS[v]  = accS[v] * __expf(accM[v] - nM) + oS * __expf(oM - nM);
            accM[v]  = nM;
        }
    }
    if (l16 == 0) {
        #pragma unroll
        for (int v = 0; v < 8; ++v) {
            s_redM[wave][v + 8 * half] = accM[v];
            s_redS[wave][v + 8 * half] = accS[v];
        }
    }
    __syncthreads();

    // ---- combine waves, fold in true logit, finalize ----
    if (tid < ROWS_PER_WG) {
        float M = -INFINITY, S = 0.0f;
        #pragma unroll
        for (int w = 0; w < WAVES; ++w) {
            float oM = s_redM[w][tid], oS = s_redS[w][tid];
            float nM = fmaxf(M, oM);
            S = S * __expf(M - nM) + oS * __expf(oM - nM);
            M = nM;
        }
        float t  = s_true[tid];
        float nM = fmaxf(M, t);
        S = S * __expf(M - nM) + __expf(t - nM);
        out[m0 + tid] = nM + __logf(S) - t;
    }
}

extern "C" void kernel_launch(void* const* d_in, const int* in_sizes, int n_in,
                              void* d_out, int out_size, void* d_ws, size_t ws_size,
                              hipStream_t stream) {
    const float* item_emb   = (const float*)d_in[0];
    const float* user_emb   = (const float*)d_in[1];
    const float* zero_bias  = (const float*)d_in[2];
    const int*   label_idx  = (const int*)d_in[3];
    const int*   sampled_id = (const int*)d_in[4];
    float* out = (float*)d_out;

    dim3 grid(BATCH / ROWS_PER_WG);   // 256 workgroups
    dim3 block(256);                  // 8 wave32 per workgroup
    hipLaunchKernelGGL(sampled_softmax_wmma, grid, block, 0, stream,
                       item_emb, user_emb, zero_bias, label_idx, sampled_id, out);
}